// SourceGCNConvEncoder_61572651155641
// MI455X (gfx1250) — compile-verified
//
#include <hip/hip_runtime.h>
#include <math.h>

typedef float v2f __attribute__((ext_vector_type(2)));
typedef float v8f __attribute__((ext_vector_type(8)));

__device__ __forceinline__ float mish_f(float x) {
    // numerically stable softplus -> tanh -> scale
    float sp = (x > 20.0f) ? x : log1pf(expf(x));
    return x * tanhf(sp);
}

// ---------------------------------------------------------------------------
// fp32 GEMM via V_WMMA_F32_16X16X4_F32 (wave32).
// C[MT*16, 256] = concat(A0[:, :kSplit], A1[:, :K-kSplit]) @ B[K, 256]
// Each wave computes a 16x64 strip (4 accumulators); block = 8 waves = 8 row
// tiles; grid.y covers 256/64 = 4 column strips.
//
// B is staged into LDS in *fragment-swizzled* form: for k-step s (4 K-values)
// and column tile jt, lane l owns the float2 {B[4s+2h][n], B[4s+2h+1][n]}
// (h = l>>4, n = l&15) at word ((s*4+jt)*32 + l)*2.  The inner loop reads each
// B fragment with a single aligned ds_load_b64 (immediate offset, no bank
// conflicts).  The A-side A0/A1 select is hoisted per 128-wide K-chunk
// (kSplit is a multiple of 128), so inner-loop A loads are one base register
// plus immediate offsets.
// ---------------------------------------------------------------------------
__global__ __launch_bounds__(256)
void wmma_gemm_split(const float* __restrict__ A0, const float* __restrict__ A1,
                     int lda0, int lda1, int kSplit,
                     const float* __restrict__ B, int ldb,
                     float* __restrict__ C, int ldc,
                     int MT, int K)
{
    __shared__ float Bs[128 * 64];  // 32 KB K-chunk of B, fragment-swizzled
    const int colBase = blockIdx.y * 64;
    const int wave = threadIdx.x >> 5;
    const int lane = threadIdx.x & 31;
    int rowTile = blockIdx.x * 8 + wave;
    if (rowTile >= MT) rowTile = MT - 1;   // wave-uniform clamp: EXEC stays all-1s
    const int m     = rowTile * 16 + (lane & 15);
    const int khalf = (lane >> 4) * 2;     // lanes 0-15: K pair {0,1}; 16-31: {2,3}

    const v2f* __restrict__ bfrag = (const v2f*)Bs;

    v8f acc0 = {}, acc1 = {}, acc2 = {}, acc3 = {};

    for (int k0 = 0; k0 < K; k0 += 128) {
        // cooperative swizzled stage of B[k0:k0+128, colBase:colBase+64]
        for (int i = threadIdx.x; i < 128 * 64; i += 256) {
            int kr = i >> 6, c = i & 63;
            float v = B[(size_t)(k0 + kr) * ldb + colBase + c];
            int s  = kr >> 2;          // k-step within chunk
            int h  = (kr >> 1) & 1;    // which half-wave holds this K pair
            int j  = kr & 1;           // low/high element of the pair
            int jt = c >> 4;           // column tile 0..3
            int n  = c & 15;
            Bs[(((s * 4 + jt) * 32) + h * 16 + n) * 2 + j] = v;
        }
        __syncthreads();

        // kSplit is a multiple of 128: the whole chunk is in A0 or in A1.
        // Hoist the base-pointer select out of the hot loop so A loads are
        // base + immediate offset (no per-step cndmask/64-bit adds).
        const float* Abase = (k0 < kSplit)
            ? (A0 + (size_t)m * lda0 + k0 + khalf)
            : (A1 + (size_t)m * lda1 + (k0 - kSplit) + khalf);

#pragma unroll 4
        for (int s = 0; s < 32; ++s) {         // 32 k-steps of depth 4
            v2f a = *(const v2f*)(Abase + s * 4);     // one global_load_b64

            v2f b0 = bfrag[(s * 4 + 0) * 32 + lane];  // ds_load_b64 each
            v2f b1 = bfrag[(s * 4 + 1) * 32 + lane];
            v2f b2 = bfrag[(s * 4 + 2) * 32 + lane];
            v2f b3 = bfrag[(s * 4 + 3) * 32 + lane];

            acc0 = __builtin_amdgcn_wmma_f32_16x16x4_f32(false, a, false, b0, (short)0, acc0, false, false);
            acc1 = __builtin_amdgcn_wmma_f32_16x16x4_f32(false, a, false, b1, (short)0, acc1, false, false);
            acc2 = __builtin_amdgcn_wmma_f32_16x16x4_f32(false, a, false, b2, (short)0, acc2, false, false);
            acc3 = __builtin_amdgcn_wmma_f32_16x16x4_f32(false, a, false, b3, (short)0, acc3, false, false);
        }
        __syncthreads();
    }

    // C/D layout: lanes 0-15 hold rows 0..7 in VGPRs 0..7, lanes 16-31 rows 8..15
    const int n     = lane & 15;
    const int mBase = rowTile * 16 + ((lane >> 4) << 3);
    float* crow = C + (size_t)mBase * ldc + colBase + n;
#pragma unroll
    for (int r = 0; r < 8; ++r) {
        crow[(size_t)r * ldc +  0] = acc0[r];
        crow[(size_t)r * ldc + 16] = acc1[r];
        crow[(size_t)r * ldc + 32] = acc2[r];
        crow[(size_t)r * ldc + 48] = acc3[r];
    }
}

// ---------------------------------------------------------------------------
// Graph kernels
// ---------------------------------------------------------------------------
__global__ void zero_kernel(float* __restrict__ p, size_t n) {
    size_t i = (size_t)blockIdx.x * 256 + threadIdx.x;
    if (i < n) p[i] = 0.0f;
}

// degA[v] = #edges with edges[0,:]==v (out-degree), degB likewise for edges[1,:]
__global__ void degrees_kernel(const int* __restrict__ edges, int E,
                               float* __restrict__ degA, float* __restrict__ degB) {
    int t = blockIdx.x * 256 + threadIdx.x;
    if (t < E) {
        atomicAdd(&degA[edges[t]], 1.0f);
        atomicAdd(&degB[edges[E + t]], 1.0f);
    }
}

// norm[e] = c / (deg_out[src] + deg_in[dst]) + 1/deg_in[dst]
// (deg_in[dst] >= 1 for any existing edge, so the where() is unconditional)
__global__ void norm_kernel(const int* __restrict__ src, const int* __restrict__ dst,
                            const float* __restrict__ degSrcOut,
                            const float* __restrict__ degDstIn,
                            const float* __restrict__ cPtr,
                            float* __restrict__ nrm, int E) {
    int t = blockIdx.x * 256 + threadIdx.x;
    if (t < E) {
        int r = src[t], c = dst[t];
        float di   = degDstIn[c];
        float dsum = degSrcOut[r] + di;
        nrm[t] = cPtr[0] * (1.0f / dsum) + 1.0f / di;
    }
}

// One wave per edge: AGG[dst] += norm * X[src]  (256 features, float4 loads)
__global__ __launch_bounds__(256)
void scatter_kernel(const int* __restrict__ src, const int* __restrict__ dst,
                    const float* __restrict__ nrm,
                    const float* __restrict__ X, float* __restrict__ AGG, int E) {
    int e = blockIdx.x * 8 + (threadIdx.x >> 5);
    if (e >= E) return;                       // wave-uniform
    int lane = threadIdx.x & 31;
    int s = src[e], d = dst[e];
    float w = nrm[e];
    const float4* xr = (const float4*)(X + (size_t)s * 256);
    float* ar = AGG + (size_t)d * 256;
#pragma unroll
    for (int c = 0; c < 2; ++c) {
        int chunk = lane + c * 32;            // 64 float4 chunks per row
        float4 v = xr[chunk];
        int base = chunk * 4;
        atomicAdd(&ar[base + 0], w * v.x);
        atomicAdd(&ar[base + 1], w * v.y);
        atomicAdd(&ar[base + 2], w * v.z);
        atomicAdd(&ar[base + 3], w * v.w);
    }
}

// v = mish(mish(v)) in place (conv's internal mish + forward's extra mish)
__global__ void mish2_kernel(float* __restrict__ v, size_t n) {
    size_t i = (size_t)blockIdx.x * 256 + threadIdx.x;
    if (i < n) v[i] = mish_f(mish_f(v[i]));
}

// out = mish(x1 + x2 + mish(mish(out)))   (out currently holds conv2 aggregate)
__global__ void final_kernel(const float* __restrict__ x1, const float* __restrict__ x2,
                             float* __restrict__ out, size_t n) {
    size_t i = (size_t)blockIdx.x * 256 + threadIdx.x;
    if (i < n) {
        float x4 = mish_f(mish_f(out[i]));
        out[i] = mish_f(x1[i] + x2[i] + x4);
    }
}

// ---------------------------------------------------------------------------
extern "C" void kernel_launch(void* const* d_in, const int* in_sizes, int n_in,
                              void* d_out, int out_size, void* d_ws, size_t ws_size,
                              hipStream_t stream) {
    const float* x  = (const float*)d_in[0];
    const int*   ei = (const int*)d_in[1];   // [2, E]: rows = ei[0..E), cols = ei[E..2E)
    const float* W1 = (const float*)d_in[2]; // [512, 256] row-major
    const float* W2 = (const float*)d_in[3]; // [512, 256] row-major
    const float* c1 = (const float*)d_in[4];
    const float* c2 = (const float*)d_in[5];
    float* out = (float*)d_out;

    const int IN_C = 512, OUT_C = 256;
    const int N = in_sizes[0] / IN_C;
    const int E = in_sizes[1] / 2;
    const size_t NC = (size_t)N * OUT_C;

    float* ws   = (float*)d_ws;
    float* x1   = ws;             // [N, 256]
    float* x2   = x1 + NC;        // [N, 256]  (also conv1 accumulator)
    float* x3   = x2 + NC;        // [N, 256]
    float* degA = x3 + NC;        // deg over ei[0,:]  (orig out-degree)
    float* degB = degA + N;       // deg over ei[1,:]  (orig in-degree)
    float* nrm  = degB + N;       // [E]

    const int MT = (N + 15) / 16;
    dim3 gemmGrid((MT + 7) / 8, OUT_C / 64);
    const unsigned ewBlocks  = (unsigned)((NC + 255) / 256);
    const unsigned eBlocks   = (unsigned)((E + 255) / 256);
    const unsigned scBlocks  = (unsigned)((E + 7) / 8);

    // --- fc1: x1 = x @ W1 ---
    wmma_gemm_split<<<gemmGrid, 256, 0, stream>>>(x, x, IN_C, IN_C, IN_C,
                                                  W1, OUT_C, x1, OUT_C, MT, IN_C);

    // --- degrees (shared by both convs; reversal just swaps roles) ---
    zero_kernel<<<(unsigned)((2 * (size_t)N + 255) / 256), 256, 0, stream>>>(degA, 2 * (size_t)N);
    degrees_kernel<<<eBlocks, 256, 0, stream>>>(ei, E, degA, degB);

    // --- conv1: src = row (ei), dst = col (ei+E) ---
    norm_kernel<<<eBlocks, 256, 0, stream>>>(ei, ei + E, degA, degB, c1, nrm, E);
    zero_kernel<<<ewBlocks, 256, 0, stream>>>(x2, NC);
    scatter_kernel<<<scBlocks, 256, 0, stream>>>(ei, ei + E, nrm, x1, x2, E);
    mish2_kernel<<<ewBlocks, 256, 0, stream>>>(x2, NC);   // x2 = mish(mish(agg1))

    // --- fc2: x3 = [x1 | x2] @ W2 ---
    wmma_gemm_split<<<gemmGrid, 256, 0, stream>>>(x1, x2, OUT_C, OUT_C, OUT_C,
                                                  W2, OUT_C, x3, OUT_C, MT, IN_C);

    // --- conv2 on reversed edges: src = col (ei+E), dst = row (ei) ---
    // deg_out' = degB (counts of ei[1,:]), deg_in' = degA (counts of ei[0,:])
    norm_kernel<<<eBlocks, 256, 0, stream>>>(ei + E, ei, degB, degA, c2, nrm, E);
    zero_kernel<<<ewBlocks, 256, 0, stream>>>(out, NC);
    scatter_kernel<<<scBlocks, 256, 0, stream>>>(ei + E, ei, nrm, x3, out, E);

    // --- out = mish(x1 + x2 + mish(mish(agg2))) ---
    final_kernel<<<ewBlocks, 256, 0, stream>>>(x1, x2, out, NC);
}